// coattn_mot_8787503087932
// MI455X (gfx1250) — compile-verified
//
#include <hip/hip_runtime.h>
#include <hip/hip_bf16.h>
#include <math.h>

typedef __attribute__((ext_vector_type(2))) float v2f;
typedef __attribute__((ext_vector_type(8))) float v8f;

namespace {
constexpr int   BB     = 4;
constexpr int   NN     = 4096;
constexpr int   MM     = 2048;
constexpr int   DD     = 256;
constexpr int   NSPLIT = 32;          // row-splits for the K^T u matvec
constexpr int   CHUNK  = NN / NSPLIT; // 128 rows per split
constexpr float FI     = 0.83333333f; // tau/(tau+reg) = 0.5/0.6
constexpr float EPSF   = 1e-16f;
}

// ---------------------------------------------------------------- init
__global__ __launch_bounds__(256) void init_kernel(float* u, float* v, unsigned* maxc) {
    int idx = blockIdx.x * 256 + threadIdx.x;
    if (idx < BB * NN) u[idx] = 1.0f / NN;
    if (idx < BB * MM) v[idx] = 1.0f / MM;
    if (idx < BB)      maxc[idx] = 0u;
}

// --------------------------------------------------- row normalize + sqnorm
// one block (256 threads == D) per row; rows flattened over (batch, row)
__global__ __launch_bounds__(256) void norm_kernel(const float* __restrict__ src,
                                                   float* __restrict__ dst,
                                                   float* __restrict__ sq) {
    __shared__ float red[256];
    const int tid = threadIdx.x;
    const size_t row = blockIdx.x;
    float val = src[row * DD + tid];
    red[tid] = val;
    __syncthreads();
    for (int o = 128; o > 0; o >>= 1) {
        if (tid < o) red[tid] = fminf(red[tid], red[tid + o]);
        __syncthreads();
    }
    float mn = red[0];
    __syncthreads();
    float nv = val - mn;
    dst[row * DD + tid] = nv;
    red[tid] = nv * nv;
    __syncthreads();
    for (int o = 128; o > 0; o >>= 1) {
        if (tid < o) red[tid] += red[tid + o];
        __syncthreads();
    }
    if (tid == 0) sq[row] = red[0];
}

// ---------------------------------------------------------- WMMA cost GEMM
// cost[i][j] = max(0, x2[i] + y2[j] - 2 * dot(xn[i], yn[j]))
// f32 WMMA 16x16x4: A 16x4 (lane&15 = row, (lane>>4)*2+{0,1} = k), B 4x16 mirrored.
// Block = 4 waves; wave w computes 16 rows, 4 tiles (64 cols) of C.
__global__ __launch_bounds__(128) void gemm_cost_kernel(const float* __restrict__ xn,
                                                        const float* __restrict__ yn,
                                                        const float* __restrict__ x2,
                                                        const float* __restrict__ y2,
                                                        float* __restrict__ Kmat,
                                                        unsigned* __restrict__ maxc) {
    const int b    = blockIdx.z;
    const int lane = threadIdx.x & 31;
    const int wave = threadIdx.x >> 5;
    const int hf   = lane >> 4;       // 0 or 1
    const int mr   = lane & 15;
    const int i0   = blockIdx.y * 64 + wave * 16;
    const int j0   = blockIdx.x * 64;

    const float* xb = xn + (size_t)b * NN * DD;
    const float* yb = yn + (size_t)b * MM * DD;

    const float* arow = xb + (size_t)(i0 + mr) * DD + 2 * hf;
    const float* brow = yb + (size_t)(j0 + mr) * DD + 2 * hf;

    v8f acc[4];
#pragma unroll
    for (int jj = 0; jj < 4; ++jj) acc[jj] = (v8f){0.f,0.f,0.f,0.f,0.f,0.f,0.f,0.f};

    for (int k = 0; k < DD; k += 4) {
        v2f a  = *(const v2f*)(arow + k);
        v2f b0 = *(const v2f*)(brow + k);
        v2f b1 = *(const v2f*)(brow + 16 * DD + k);
        v2f b2 = *(const v2f*)(brow + 32 * DD + k);
        v2f b3 = *(const v2f*)(brow + 48 * DD + k);
        acc[0] = __builtin_amdgcn_wmma_f32_16x16x4_f32(false, a, false, b0, (short)0, acc[0], false, false);
        acc[1] = __builtin_amdgcn_wmma_f32_16x16x4_f32(false, a, false, b1, (short)0, acc[1], false, false);
        acc[2] = __builtin_amdgcn_wmma_f32_16x16x4_f32(false, a, false, b2, (short)0, acc[2], false, false);
        acc[3] = __builtin_amdgcn_wmma_f32_16x16x4_f32(false, a, false, b3, (short)0, acc[3], false, false);
    }

    const float* x2b = x2 + (size_t)b * NN;
    const float* y2b = y2 + (size_t)b * MM;
    float y2v[4];
#pragma unroll
    for (int jj = 0; jj < 4; ++jj) y2v[jj] = y2b[j0 + jj * 16 + mr];

    float mx = 0.0f;
#pragma unroll
    for (int r = 0; r < 8; ++r) {
        const int i = i0 + r + 8 * hf;          // C layout: VGPR r -> row r (+8 for upper lanes)
        const float xs = x2b[i];
#pragma unroll
        for (int jj = 0; jj < 4; ++jj) {
            const int j = j0 + jj * 16 + mr;
            float c = xs + y2v[jj] - 2.0f * acc[jj][r];
            c = fmaxf(c, 0.0f);
            mx = fmaxf(mx, c);
            Kmat[((size_t)b * NN + i) * MM + j] = c;
        }
    }
    // wave max -> per-batch atomic max (uint compare valid: values >= 0)
    for (int off = 16; off > 0; off >>= 1)
        mx = fmaxf(mx, __shfl_xor(mx, off, 32));
    if (lane == 0) atomicMax(&maxc[b], __float_as_uint(mx));
}

// ------------------------------------------------------------- K = exp(-Mc/reg)
__global__ __launch_bounds__(256) void expk_kernel(float* __restrict__ Kmat,
                                                   const unsigned* __restrict__ maxc) {
    const size_t per4 = (size_t)NN * MM / 4;
    const size_t stride = (size_t)gridDim.x * 256;
    for (int b = 0; b < BB; ++b) {
        const float inv = 1.0f / (0.1f * __uint_as_float(maxc[b]));
        float4* Kb = (float4*)Kmat + (size_t)b * per4;
        for (size_t i = (size_t)blockIdx.x * 256 + threadIdx.x; i < per4; i += stride) {
            float4 c = Kb[i];
            c.x = expf(-c.x * inv);
            c.y = expf(-c.y * inv);
            c.z = expf(-c.z * inv);
            c.w = expf(-c.w * inv);
            Kb[i] = c;
        }
    }
}

// ------------------------------------------- u[i] = (a / (K v + eps))^fi
// wave per row, float4 (b128) loads, v staged in LDS (shared by 8 waves)
__global__ __launch_bounds__(256) void rowmat_u_kernel(const float* __restrict__ Kmat,
                                                       const float* __restrict__ v,
                                                       float* __restrict__ u) {
    __shared__ float sv[MM];
    const int b = blockIdx.y;
    for (int j = threadIdx.x; j < MM; j += 256) sv[j] = v[(size_t)b * MM + j];
    __syncthreads();
    const int lane = threadIdx.x & 31;
    const int wave = threadIdx.x >> 5;
    const int i    = blockIdx.x * 8 + wave;
    const float4* Kr  = (const float4*)(Kmat + ((size_t)b * NN + i) * MM);
    const float4* sv4 = (const float4*)sv;
    float s = 0.0f;
#pragma unroll 4
    for (int j = lane; j < MM / 4; j += 32) {
        float4 k4 = Kr[j];
        float4 v4 = sv4[j];
        s += k4.x * v4.x + k4.y * v4.y + k4.z * v4.z + k4.w * v4.w;
    }
    for (int off = 16; off > 0; off >>= 1) s += __shfl_xor(s, off, 32);
    if (lane == 0)
        u[(size_t)b * NN + i] = powf((1.0f / NN) / (s + EPSF), FI);
}

// ---------------------------- partial K^T u: 256 blocks (split rows for occupancy)
// block = (1024 cols as float4) x (CHUNK rows); part[b][split][j] partial sums
__global__ __launch_bounds__(256) void colmat_part_kernel(const float* __restrict__ Kmat,
                                                          const float* __restrict__ u,
                                                          float* __restrict__ part) {
    __shared__ float su[CHUNK];
    const int b  = blockIdx.z;
    const int s  = blockIdx.y;
    const int i0 = s * CHUNK;
    if (threadIdx.x < CHUNK) su[threadIdx.x] = u[(size_t)b * NN + i0 + threadIdx.x];
    __syncthreads();
    const int j4 = blockIdx.x * 256 + threadIdx.x;             // float4 column index
    const float4* Kb = (const float4*)(Kmat + ((size_t)b * NN + i0) * MM);
    float4 acc = make_float4(0.f, 0.f, 0.f, 0.f);
#pragma unroll 4
    for (int i = 0; i < CHUNK; ++i) {
        float4 k4 = Kb[(size_t)i * (MM / 4) + j4];
        const float ui = su[i];
        acc.x += k4.x * ui;
        acc.y += k4.y * ui;
        acc.z += k4.z * ui;
        acc.w += k4.w * ui;
    }
    ((float4*)part)[((size_t)b * NSPLIT + s) * (MM / 4) + j4] = acc;
}

// ------------------------------- v[j] = (b / (sum_s part + eps))^fi
__global__ __launch_bounds__(256) void colmat_fin_kernel(const float* __restrict__ part,
                                                         float* __restrict__ v) {
    const int idx = blockIdx.x * 256 + threadIdx.x;  // 0 .. B*M-1
    const int b = idx / MM;
    const int j = idx - b * MM;
    float s = 0.0f;
#pragma unroll
    for (int t = 0; t < NSPLIT; ++t)
        s += part[((size_t)b * NSPLIT + t) * MM + j];
    v[idx] = powf((1.0f / MM) / (s + EPSF), FI);
}

// ------------------------- out[b][m][n] = u[n] * K[n][m] * v[m]  (LDS transpose)
__global__ __launch_bounds__(256) void plan_t_kernel(const float* __restrict__ Kmat,
                                                     const float* __restrict__ u,
                                                     const float* __restrict__ v,
                                                     float* __restrict__ out) {
    __shared__ float tile[32 * 33];
    const int b  = blockIdx.z;
    const int n0 = blockIdx.x * 32;
    const int m0 = blockIdx.y * 32;
    const int tx = threadIdx.x;   // 0..31
    const int ty = threadIdx.y;   // 0..7
    const float* ub = u + (size_t)b * NN;
    const float* vb = v + (size_t)b * MM;
    const float vj = vb[m0 + tx];
#pragma unroll
    for (int r = 0; r < 4; ++r) {
        const int n = n0 + ty + r * 8;
        float val = Kmat[((size_t)b * NN + n) * MM + (m0 + tx)] * ub[n] * vj;
        tile[tx * 33 + (ty + r * 8)] = val;   // [m_local][n_local]
    }
    __syncthreads();
#pragma unroll
    for (int r = 0; r < 4; ++r) {
        const int m = m0 + ty + r * 8;
        out[((size_t)b * MM + m) * NN + (n0 + tx)] = tile[(ty + r * 8) * 33 + tx];
    }
}

// ================================================================= launch
extern "C" void kernel_launch(void* const* d_in, const int* in_sizes, int n_in,
                              void* d_out, int out_size, void* d_ws, size_t ws_size,
                              hipStream_t stream) {
    const float* x = (const float*)d_in[0];   // (B, N, D)
    const float* y = (const float*)d_in[1];   // (B, M, D)
    float* out = (float*)d_out;               // (B, M, N)

    // workspace carve-up (floats)
    float* ws   = (float*)d_ws;
    float* Kmat = ws;                                   // B*N*M  (cost, then Gibbs kernel)
    float* xn   = Kmat + (size_t)BB * NN * MM;          // B*N*D
    float* yn   = xn   + (size_t)BB * NN * DD;          // B*M*D
    float* x2   = yn   + (size_t)BB * MM * DD;          // B*N
    float* y2   = x2   + (size_t)BB * NN;               // B*M
    float* u    = y2   + (size_t)BB * MM;               // B*N
    float* v    = u    + (size_t)BB * NN;               // B*M
    unsigned* maxc = (unsigned*)(v + (size_t)BB * MM);  // B (padded to 64)
    float* part = (float*)(maxc + 64);                  // B*NSPLIT*M  (1 MB)

    // 1. init u, v, per-batch max
    init_kernel<<<(BB * NN + 255) / 256, 256, 0, stream>>>(u, v, maxc);

    // 2. normalize features + squared row norms
    norm_kernel<<<BB * NN, 256, 0, stream>>>(x, xn, x2);
    norm_kernel<<<BB * MM, 256, 0, stream>>>(y, yn, y2);

    // 3. WMMA cost GEMM + per-batch max
    {
        dim3 grid(MM / 64, NN / 64, BB);
        gemm_cost_kernel<<<grid, 128, 0, stream>>>(xn, yn, x2, y2, Kmat, maxc);
    }

    // 4. Gibbs kernel in place
    expk_kernel<<<4096, 256, 0, stream>>>(Kmat, maxc);

    // 5. 100 Sinkhorn iterations (K stays L2-resident: 128 MB < 192 MB L2)
    for (int it = 0; it < 100; ++it) {
        dim3 gu(NN / 8, BB);
        rowmat_u_kernel<<<gu, 256, 0, stream>>>(Kmat, v, u);
        dim3 gp(MM / 1024, NSPLIT, BB);                 // 256 blocks
        colmat_part_kernel<<<gp, 256, 0, stream>>>(Kmat, u, part);
        colmat_fin_kernel<<<(BB * MM) / 256, 256, 0, stream>>>(part, v);
    }

    // 6. transposed transport plan
    {
        dim3 grid(NN / 32, MM / 32, BB);
        dim3 blk(32, 8);
        plan_t_kernel<<<grid, blk, 0, stream>>>(Kmat, u, v, out);
    }
}